// HGNN_68092411511615
// MI455X (gfx1250) — compile-verified
//
#include <hip/hip_runtime.h>
#include <cstddef>
#include <cstdint>

// ---------------------------------------------------------------------------
// HGCN (Poincare ball, c=1) 2-layer forward for MI455X (gfx1250).
// Bandwidth/atomic bound; GEMM (13 GFLOP fp32) via V_WMMA_F32_16X16X4_F32.
// W staged to LDS with GLOBAL_LOAD_ASYNC_TO_LDS_B128 (ASYNCcnt-tracked).
// ---------------------------------------------------------------------------

#define MIN_NORM 1e-15f
#define MAXN     0.996f     // (1 - 4e-3)/sqrt(c), c = 1

typedef float v2f __attribute__((ext_vector_type(2)));
typedef float v8f __attribute__((ext_vector_type(8)));
typedef int   v4i_vs __attribute__((vector_size(16)));           // int4 (gcc-style)
typedef __attribute__((address_space(1))) v4i_vs as1_v4i;        // global int4
typedef __attribute__((address_space(3))) v4i_vs as3_v4i;        // LDS int4

__device__ __forceinline__ float wsum(float v) {
#pragma unroll
  for (int o = 16; o > 0; o >>= 1) v += __shfl_xor(v, o, 32);
  return v;
}
__device__ __forceinline__ float dot4(float4 a, float4 b) {
  return a.x * b.x + a.y * b.y + a.z * b.z + a.w * b.w;
}
__device__ __forceinline__ float4 mul4(float4 a, float s) {
  return make_float4(a.x * s, a.y * s, a.z * s, a.w * s);
}
__device__ __forceinline__ float artanh_c(float z) {   // z >= 0 here
  z = fminf(z, 1.0f - 1e-7f);
  return 0.5f * logf((1.0f + z) / (1.0f - z));
}

// ---------------------------------------------------------------------------
// bias_prep: hb = proj(expmap0(b)) for the 4 bias vectors (blin1,b1,blin2,b2).
// ---------------------------------------------------------------------------
__global__ void bias_prep(const float* blin1, const float* b1,
                          const float* blin2, const float* b2, float* hb) {
  const int w = threadIdx.x >> 5, lane = threadIdx.x & 31;
  const float* v = (w == 0) ? blin1 : (w == 1) ? b1 : (w == 2) ? blin2 : b2;
  float4 u = ((const float4*)v)[lane];
  float bn = fmaxf(sqrtf(wsum(dot4(u, u))), MIN_NORM);
  float4 e = mul4(u, tanhf(bn) / bn);                  // expmap0
  float en = fmaxf(sqrtf(wsum(dot4(e, e))), MIN_NORM); // proj
  if (en > MAXN) e = mul4(e, MAXN / en);
  ((float4*)hb)[(size_t)w * 32 + lane] = e;
}

// ---------------------------------------------------------------------------
// gemm128_wmma: MX = X @ W^T, d = 128. 8 waves/block, 16-row tile per wave,
// 8 f32 WMMA accumulators x 32 K-steps. W in LDS, row stride padded to 132
// floats (132 mod 64 = 4 banks/lane -> conflict-free ds_read_b64).
// ---------------------------------------------------------------------------
#define LDSW 132
__global__ __launch_bounds__(256) void gemm128_wmma(
    const float* __restrict__ X, const float* __restrict__ W,
    float* __restrict__ MX, int N) {
  __shared__ float lW[128 * LDSW];
  const int tid = threadIdx.x;

#if __has_builtin(__builtin_amdgcn_global_load_async_to_lds_b128)
  // gfx1250 async direct global->LDS copy (ASYNCcnt-tracked), 16B per lane.
#pragma unroll
  for (int it = 0; it < 16; ++it) {          // 4096 float4 = 128x128 W
    int idx4 = tid + it * 256;
    int j = idx4 >> 5;                       // W row
    int k4 = (idx4 & 31) << 2;               // col group of 4
    as1_v4i* g = (as1_v4i*)(uintptr_t)(const void*)(W + (size_t)idx4 * 4);
    as3_v4i* l = (as3_v4i*)(uintptr_t)(void*)&lW[j * LDSW + k4];
    __builtin_amdgcn_global_load_async_to_lds_b128(g, l, 0, 0);
  }
#if __has_builtin(__builtin_amdgcn_s_wait_asynccnt)
  __builtin_amdgcn_s_wait_asynccnt(0);
#else
  asm volatile("s_wait_asynccnt 0x0" ::: "memory");
#endif
#else
  // Fallback: synchronous staging through VGPRs.
#pragma unroll
  for (int it = 0; it < 16; ++it) {
    int idx4 = tid + it * 256;
    int j = idx4 >> 5;
    int k4 = (idx4 & 31) << 2;
    float4 w = ((const float4*)W)[idx4];
    float* p = &lW[j * LDSW + k4];
    p[0] = w.x; p[1] = w.y; p[2] = w.z; p[3] = w.w;
  }
#endif
  __syncthreads();

  const int wave = tid >> 5;
  const int lane = tid & 31;
  const int hi = lane >> 4;                  // half-wave select (K pair)
  const int ln = lane & 15;                  // M (for A) / N (for B)
  const int rbase = blockIdx.x * 128 + wave * 16;
  int rowA = rbase + ln;
  if (rowA >= N) rowA = N - 1;               // clamp, keep EXEC all-ones
  const float* xrow = X + (size_t)rowA * 128 + 2 * hi;

  v8f acc[8];
  const v8f vzero = {0.f, 0.f, 0.f, 0.f, 0.f, 0.f, 0.f, 0.f};
#pragma unroll
  for (int nt = 0; nt < 8; ++nt) acc[nt] = vzero;

  for (int k0 = 0; k0 < 128; k0 += 4) {
    // A fragment: 16x4 f32. lane<16: (M=ln, K=k0,k0+1); lane>=16: K=k0+2,k0+3
    v2f a; a.x = xrow[k0]; a.y = xrow[k0 + 1];
#pragma unroll
    for (int nt = 0; nt < 8; ++nt) {
      // B fragment: 4x16 f32, B[k][n] = W[n0+n][k0+k]; mirrored lane layout.
      const float* wp = &lW[(nt * 16 + ln) * LDSW + k0 + 2 * hi];
      v2f b; b.x = wp[0]; b.y = wp[1];
      acc[nt] = __builtin_amdgcn_wmma_f32_16x16x4_f32(
          false, a, false, b, (short)0, acc[nt], false, false);
    }
  }
  // C/D layout: VGPR v -> row (v + 8*hi), col = nt*16 + ln
#pragma unroll
  for (int nt = 0; nt < 8; ++nt) {
    const int col = nt * 16 + ln;
#pragma unroll
    for (int i = 0; i < 8; ++i) {
      int gr = rbase + i + 8 * hi;
      if (gr < N) MX[(size_t)gr * 128 + col] = acc[nt][i];
    }
  }
}

// ---------------------------------------------------------------------------
// row_pre: tail of mobius_matvec + proj + mobius_add(hb_lin) + proj + logmap0.
// One wave per node row; t is written IN PLACE over MX, and also into AGG as
// the self-loop initializer for the segment sum.
// ---------------------------------------------------------------------------
__global__ void row_pre(const float* __restrict__ X, float* MX_T,
                        const float* __restrict__ hb, float* __restrict__ AGG,
                        int N) {
  const int gid = blockIdx.x * blockDim.x + threadIdx.x;
  const int row = gid >> 5;
  if (row >= N) return;
  const int lane = gid & 31;
  const size_t ri = (size_t)row * 32 + lane;

  float4 xv = ((const float4*)X)[ri];
  float4 m  = ((const float4*)MX_T)[ri];
  float x2  = wsum(dot4(xv, xv));
  float mx2 = wsum(dot4(m, m));
  float xn  = fmaxf(sqrtf(x2), MIN_NORM);
  float mxn = fmaxf(sqrtf(mx2), MIN_NORM);
  float s = tanhf(mxn / xn * artanh_c(xn)) / mxn;      // mobius_matvec scale
  if (mx2 == 0.f) s = 0.f;                             // all-zero mx -> 0
  float4 r = mul4(m, s);
  float rn = fmaxf(sqrtf(wsum(dot4(r, r))), MIN_NORM); // proj
  if (rn > MAXN) r = mul4(r, MAXN / rn);

  float4 hv = ((const float4*)hb)[lane];               // mobius_add(r, hb)
  float r2 = wsum(dot4(r, r));
  float y2 = wsum(dot4(hv, hv));
  float xy = wsum(dot4(r, hv));
  float na = 1.f + 2.f * xy + y2;
  float nb = 1.f - r2;
  float den = fmaxf(1.f + 2.f * xy + r2 * y2, MIN_NORM);
  float4 h = make_float4((na * r.x + nb * hv.x) / den, (na * r.y + nb * hv.y) / den,
                         (na * r.z + nb * hv.z) / den, (na * r.w + nb * hv.w) / den);
  float hn = fmaxf(sqrtf(wsum(dot4(h, h))), MIN_NORM); // proj
  if (hn > MAXN) { h = mul4(h, MAXN / hn); hn = MAXN; }

  float4 t = mul4(h, artanh_c(hn) / hn);               // logmap0
  ((float4*)MX_T)[ri] = t;                             // t (in place)
  ((float4*)AGG)[ri]  = t;                             // self-loop init
}

// ---------------------------------------------------------------------------
// scatter_add: AGG[dst] += T[src] per edge. One wave per edge; lane l owns
// columns l, l+32, l+64, l+96 so every load and every atomic instruction is
// one contiguous 128B line (best-case combining at the L2 atomic units).
// ---------------------------------------------------------------------------
__global__ void scatter_add(const float* __restrict__ T,
                            const int* __restrict__ src,
                            const int* __restrict__ dst,
                            float* __restrict__ AGG, int E) {
  const int gid = blockIdx.x * blockDim.x + threadIdx.x;
  const int e = gid >> 5;
  if (e >= E) return;
  const int lane = gid & 31;
  const int s = src[e];
  const int d = dst[e];
  const float* tr = T + (size_t)s * 128 + lane;
  float* ar = AGG + (size_t)d * 128 + lane;
#pragma unroll
  for (int j = 0; j < 4; ++j) {
    unsafeAtomicAdd(ar + j * 32, tr[j * 32]);
  }
}

// ---------------------------------------------------------------------------
// row_post: proj(expmap0(agg)) then mobius_add(hb_out) + proj.
// Safe for OUT == AGG (row-local, loads precede stores).
// ---------------------------------------------------------------------------
__global__ void row_post(const float* AGG, const float* __restrict__ hb,
                         float* OUT, int N) {
  const int gid = blockIdx.x * blockDim.x + threadIdx.x;
  const int row = gid >> 5;
  if (row >= N) return;
  const int lane = gid & 31;
  const size_t ri = (size_t)row * 32 + lane;

  float4 u = ((const float4*)AGG)[ri];
  float un = fmaxf(sqrtf(wsum(dot4(u, u))), MIN_NORM);
  float4 e = mul4(u, tanhf(un) / un);                  // expmap0
  float en = fmaxf(sqrtf(wsum(dot4(e, e))), MIN_NORM); // proj
  if (en > MAXN) e = mul4(e, MAXN / en);

  float4 hv = ((const float4*)hb)[lane];               // mobius_add(e, hb)
  float e2 = wsum(dot4(e, e));
  float y2 = wsum(dot4(hv, hv));
  float xy = wsum(dot4(e, hv));
  float na = 1.f + 2.f * xy + y2;
  float nb = 1.f - e2;
  float den = fmaxf(1.f + 2.f * xy + e2 * y2, MIN_NORM);
  float4 h = make_float4((na * e.x + nb * hv.x) / den, (na * e.y + nb * hv.y) / den,
                         (na * e.z + nb * hv.z) / den, (na * e.w + nb * hv.w) / den);
  float hn = fmaxf(sqrtf(wsum(dot4(h, h))), MIN_NORM); // proj
  if (hn > MAXN) h = mul4(h, MAXN / hn);
  ((float4*)OUT)[ri] = h;
}

// ---------------------------------------------------------------------------
extern "C" void kernel_launch(void* const* d_in, const int* in_sizes, int n_in,
                              void* d_out, int out_size, void* d_ws,
                              size_t ws_size, hipStream_t stream) {
  const float* x     = (const float*)d_in[0];
  const int*   ei    = (const int*)  d_in[1];
  const float* W1    = (const float*)d_in[2];
  const float* blin1 = (const float*)d_in[3];
  const float* b1    = (const float*)d_in[4];
  const float* W2    = (const float*)d_in[5];
  const float* blin2 = (const float*)d_in[6];
  const float* b2    = (const float*)d_in[7];

  const int N = in_sizes[0] / 128;
  const int E = in_sizes[1] / 2;
  const int* src = ei;
  const int* dst = ei + E;

  float* out  = (float*)d_out;
  float* bufA = (float*)d_ws;                 // N*128
  float* bufB = bufA + (size_t)N * 128;       // N*128
  float* hb   = bufB + (size_t)N * 128;       // 4*128 bias vectors

  const dim3 blk(256);
  const int gGemm = (N + 127) / 128;
  const int gRow  = (int)(((size_t)N * 32 + 255) / 256);
  const int gEdge = (int)(((size_t)E * 32 + 255) / 256);

  bias_prep<<<1, 128, 0, stream>>>(blin1, b1, blin2, b2, hb);

  // ---- Layer 1 ----  MX1 -> bufA (t in place), AGG1 = bufB, H -> bufA
  gemm128_wmma<<<gGemm, blk, 0, stream>>>(x, W1, bufA, N);
  row_pre<<<gRow, blk, 0, stream>>>(x, bufA, hb + 0, bufB, N);
  scatter_add<<<gEdge, blk, 0, stream>>>(bufA, src, dst, bufB, E);
  row_post<<<gRow, blk, 0, stream>>>(bufB, hb + 128, bufA, N);

  // ---- Layer 2 ----  MX2 -> bufB (t in place), AGG2 = d_out (in place post)
  gemm128_wmma<<<gGemm, blk, 0, stream>>>(bufA, W2, bufB, N);
  row_pre<<<gRow, blk, 0, stream>>>(bufA, bufB, hb + 256, out, N);
  scatter_add<<<gEdge, blk, 0, stream>>>(bufB, src, dst, out, E);
  row_post<<<gRow, blk, 0, stream>>>(out, hb + 384, out, N);
}